// GCN_18949395709960
// MI455X (gfx1250) — compile-verified
//
#include <hip/hip_runtime.h>
#include <hip/hip_bf16.h>

#define GCN_N      100000
#define GCN_IN_FT  256
#define GCN_OUT_FT 128
#define GCN_E      1600000

typedef float v2f __attribute__((ext_vector_type(2)));
typedef float v8f __attribute__((ext_vector_type(8)));

// ---------------------------------------------------------------------------
// Phase 1: fts[n,o] = sum_d seq[n,d] * W[o,d]   via V_WMMA_F32_16X16X4_F32
// One wave computes a 16-row x 128-col output stripe: 8 accumulator tiles,
// one A-fragment load per k-step shared across 8 WMMAs. seq is read exactly
// once from HBM (102.4 MB); W (128 KB) stays cache-resident.
// A 16x4 layout : lane half h holds K = 2h, 2h+1 in its two VGPRs (M = lane%16)
// B 4x16 layout : row-striped across lanes (N = lane%16), K mirrors A
// C/D 16x16     : VGPR v holds M = v + 8h, N = lane%16
// ---------------------------------------------------------------------------
__global__ __launch_bounds__(32) void gcn_gemm_wmma(
    const float* __restrict__ seq,   // [N, IN_FT]
    const float* __restrict__ W,     // [OUT_FT, IN_FT]
    float* __restrict__ fts)         // [N, OUT_FT]
{
    const int lane = threadIdx.x;        // 0..31
    const int h    = lane >> 4;          // lane half (0/1)
    const int m    = lane & 15;
    const int r0   = blockIdx.x * 16;    // row tile base (N dim)

    const float* arow  = seq + (size_t)(r0 + m) * GCN_IN_FT + 2 * h;
    // B[k][n] = W[n][k]; col tile t adds a constant t*16*IN_FT float offset
    const float* bbase = W + (size_t)m * GCN_IN_FT + 2 * h;

    v8f acc[8];
#pragma unroll
    for (int t = 0; t < 8; ++t) acc[t] = (v8f){};

#pragma unroll 2
    for (int k = 0; k < GCN_IN_FT; k += 4) {
        const v2f a = *(const v2f*)(arow + k);
#pragma unroll
        for (int t = 0; t < 8; ++t) {
            const v2f b = *(const v2f*)(bbase + k + t * (16 * GCN_IN_FT));
            acc[t] = __builtin_amdgcn_wmma_f32_16x16x4_f32(
                /*neg_a=*/false, a, /*neg_b=*/false, b,
                /*c_mod=*/(short)0, acc[t], /*reuse_a=*/false, /*reuse_b=*/false);
        }
    }

    float* out = fts + (size_t)r0 * GCN_OUT_FT + m;
#pragma unroll
    for (int t = 0; t < 8; ++t)
#pragma unroll
        for (int v = 0; v < 8; ++v)
            out[(size_t)(v + 8 * h) * GCN_OUT_FT + 16 * t] = acc[t][v];
}

// ---------------------------------------------------------------------------
// Phase 2: agg[dst] += val * fts[src]   (hardware fp32 atomics)
// 32 lanes per edge; edge index forced uniform via readfirstlane so the
// edge_src/dst/val reads become scalar loads. Each lane gathers one float4
// (coalesced 512B per edge) and issues 4 global_atomic_add_f32.
// ---------------------------------------------------------------------------
__global__ __launch_bounds__(256) void gcn_scatter(
    const float4* __restrict__ fts4,      // [N, OUT_FT/4]
    const int*    __restrict__ edge_src,
    const int*    __restrict__ edge_dst,
    const float*  __restrict__ edge_val,
    float*        __restrict__ agg)       // [N, OUT_FT]
{
    // One wave per edge: wave id within block is uniform across its 32 lanes.
    const int wave_in_blk = __builtin_amdgcn_readfirstlane(threadIdx.x >> 5);
    const int e  = blockIdx.x * 8 + wave_in_blk;
    if (e >= GCN_E) return;
    const int q4 = threadIdx.x & 31;      // which float4 of the row (0..31)

    const int   s = edge_src[e];
    const int   d = edge_dst[e];
    const float v = edge_val[e];

    const float4 msg = fts4[(size_t)s * (GCN_OUT_FT / 4) + q4];
    float* dp = agg + (size_t)d * GCN_OUT_FT + q4 * 4;

    unsafeAtomicAdd(dp + 0, msg.x * v);
    unsafeAtomicAdd(dp + 1, msg.y * v);
    unsafeAtomicAdd(dp + 2, msg.z * v);
    unsafeAtomicAdd(dp + 3, msg.w * v);
}

// ---------------------------------------------------------------------------
// Phase 3: out = prelu(agg + bias), float4-vectorized.
// ---------------------------------------------------------------------------
__global__ __launch_bounds__(256) void gcn_finalize(
    const float4* __restrict__ agg4,
    const float4* __restrict__ bias4,     // [OUT_FT/4]
    const float*  __restrict__ alpha_p,
    float4*       __restrict__ out4)
{
    const long long i = (long long)blockIdx.x * blockDim.x + threadIdx.x;
    if (i >= (long long)GCN_N * (GCN_OUT_FT / 4)) return;

    const float alpha = *alpha_p;
    const float4 a = agg4[i];
    const float4 b = bias4[i & ((GCN_OUT_FT / 4) - 1)];

    float4 o;
    o.x = a.x + b.x; o.y = a.y + b.y; o.z = a.z + b.z; o.w = a.w + b.w;
    o.x = (o.x >= 0.0f) ? o.x : alpha * o.x;
    o.y = (o.y >= 0.0f) ? o.y : alpha * o.y;
    o.z = (o.z >= 0.0f) ? o.z : alpha * o.z;
    o.w = (o.w >= 0.0f) ? o.w : alpha * o.w;
    out4[i] = o;
}

extern "C" void kernel_launch(void* const* d_in, const int* in_sizes, int n_in,
                              void* d_out, int out_size, void* d_ws, size_t ws_size,
                              hipStream_t stream) {
    const float* seq      = (const float*)d_in[0];
    const int*   edge_src = (const int*)d_in[1];
    const int*   edge_dst = (const int*)d_in[2];
    const float* edge_val = (const float*)d_in[3];
    const float* W        = (const float*)d_in[4];
    const float* bias     = (const float*)d_in[5];
    const float* alpha    = (const float*)d_in[6];

    float* out = (float*)d_out;           // doubles as fts scratch in phase 1/2
    float* agg = (float*)d_ws;            // N * OUT_FT floats (51.2 MB)

    const size_t agg_bytes = (size_t)GCN_N * GCN_OUT_FT * sizeof(float);

    // Zero the aggregation buffer (stream-ordered; graph-capture safe).
    hipMemsetAsync(agg, 0, agg_bytes, stream);

    // Phase 1: WMMA GEMM -> fts (stored in d_out). One wave per 16-row stripe.
    gcn_gemm_wmma<<<GCN_N / 16, 32, 0, stream>>>(seq, W, out);

    // Phase 2: edge scatter with fp32 hardware atomics into agg (8 edges/block).
    const int sblocks = GCN_E / 8;        // 200000
    gcn_scatter<<<sblocks, 256, 0, stream>>>((const float4*)out, edge_src,
                                             edge_dst, edge_val, agg);

    // Phase 3: bias + PReLU -> d_out.
    const long long fthreads = (long long)GCN_N * (GCN_OUT_FT / 4);
    const int fblocks = (int)((fthreads + 255) / 256);  // 12500
    gcn_finalize<<<fblocks, 256, 0, stream>>>((const float4*)agg,
                                              (const float4*)bias, alpha,
                                              (float4*)d_out);
}